// GCNLayer_52682068853352
// MI455X (gfx1250) — compile-verified
//
#include <hip/hip_runtime.h>

// ---------------------------------------------------------------------------
// GCN layer for MI455X (gfx1250, wave32):
//   agg = segment_sum(a_values * Xf[cols], rows)   [SpMM via f32 atomics in L2]
//   out = agg.reshape(200000,100) @ W(100,100)     [V_WMMA_F32_16X16X4_F32]
// Roofline: GEMM is 4 GFLOP over ~160MB -> memory bound; SpMM is gather/atomic
// bound with X (80MB) + agg (80MB) resident in the 192MB L2.
// ---------------------------------------------------------------------------

typedef __attribute__((ext_vector_type(2))) float v2f;
typedef __attribute__((ext_vector_type(8))) float v8f;

#define NNODES 50000
#define NNZ_E  500000
#define BDIM   4
#define FDIM   100
#define ODIM   100
#define MROWS  (NNODES * BDIM)   // 200000 GEMM rows
#define KD     FDIM              // 100
#define ND     ODIM              // 100
#define NT     7                 // ceil(100/16) N-tiles
#define CHUNKS 100               // (B*F)/4 float4 chunks per node row

// ---- clear agg scratch (80 MB) --------------------------------------------
__global__ void zero_ws_kernel(float4* __restrict__ p, long long n4) {
  long long t = blockIdx.x * (long long)blockDim.x + threadIdx.x;
  if (t < n4) p[t] = make_float4(0.f, 0.f, 0.f, 0.f);
}

// ---- SpMM scatter: one thread per (edge, float4 chunk) --------------------
__global__ void spmm_scatter_kernel(const int* __restrict__ rows,
                                    const int* __restrict__ cols,
                                    const float* __restrict__ vals,
                                    const float4* __restrict__ X4,
                                    float* __restrict__ agg) {
  long long t = blockIdx.x * (long long)blockDim.x + threadIdx.x;
  const long long total = (long long)NNZ_E * CHUNKS;
  if (t >= total) return;
  const int e = (int)(t / CHUNKS);
  const int c = (int)(t % CHUNKS);
  const int r   = rows[e];
  const int src = cols[e];
  const float v = vals[e];
  const float4 x = X4[(long long)src * CHUNKS + c];
  float* dst = agg + (long long)r * (BDIM * FDIM) + c * 4;
  atomicAdd(dst + 0, v * x.x);
  atomicAdd(dst + 1, v * x.y);
  atomicAdd(dst + 2, v * x.z);
  atomicAdd(dst + 3, v * x.w);
}

// ---- GEMM: one wave computes a 16x100 output strip with f32 WMMA ----------
// A (16x4 f32, v2f/lane):  lanes 0-15 hold M=lane, K={kk,kk+1};
//                          lanes 16-31 hold M=lane-16, K={kk+2,kk+3}.
// B (4x16 f32, v2f/lane):  VGPR j -> row K=kk+j (lanes 0-15) / K=kk+j+2
//                          (lanes 16-31), column N = n0 + (lane&15).
// C (16x16 f32, v8f):      c[r] -> M = r + 8*(lane>>4), N = lane&15.
__global__ __launch_bounds__(128) void gemm_wmma_kernel(
    const float* __restrict__ agg, const float* __restrict__ W,
    float* __restrict__ out) {
  __shared__ float sW[KD * ND];  // 40 KB of the 320 KB WGP LDS
  for (int i = threadIdx.x; i < KD * ND; i += 128) sW[i] = W[i];
  __syncthreads();

  const int lane = threadIdx.x & 31;
  const int wv   = (blockIdx.x << 2) | (threadIdx.x >> 5);  // wave id
  const int row0 = wv << 4;                                 // 16 rows / wave
  const int mL   = lane & 15;
  const int half = lane >> 4;       // 0 or 1
  const int kOff = half << 1;       // A/B lane K offset: 0 or 2
  const int nL   = lane & 15;

  v8f acc[NT];
#pragma unroll
  for (int t = 0; t < NT; ++t) acc[t] = (v8f){0.f,0.f,0.f,0.f,0.f,0.f,0.f,0.f};

  // partial last N-tile (n0 = 96): mask invalid B columns to zero (selects
  // only -- EXEC stays all-ones as WMMA requires)
  const float lastMask = (96 + nL < ND) ? 1.0f : 0.0f;
  const int   lastCol  = (96 + nL < ND) ? (96 + nL) : (ND - 1);

  const float* aBase = agg + (long long)(row0 + mL) * KD;

  for (int kk = 0; kk < KD; kk += 4) {
    const float2 av = *(const float2*)(aBase + kk + kOff);  // 8B aligned
    v2f a; a.x = av.x; a.y = av.y;

    const float* wr0 = sW + (kk + kOff) * ND;  // B row K = kk+kOff
    const float* wr1 = wr0 + ND;               // B row K = kk+kOff+1
#pragma unroll
    for (int t = 0; t < NT; ++t) {
      const int   col = (t < 6) ? (t * 16 + nL) : lastCol;
      const float msk = (t < 6) ? 1.0f : lastMask;
      v2f b;
      b.x = wr0[col] * msk;
      b.y = wr1[col] * msk;
      acc[t] = __builtin_amdgcn_wmma_f32_16x16x4_f32(
          /*neg_a=*/false, a, /*neg_b=*/false, b,
          /*c_mod=*/(short)0, acc[t], /*reuse_a=*/false, /*reuse_b=*/false);
    }
  }

  const int mS = half << 3;  // upper half-wave writes rows M=8..15
#pragma unroll
  for (int t = 0; t < NT; ++t) {
    const int n = t * 16 + nL;
    if (n < ND) {
#pragma unroll
      for (int r = 0; r < 8; ++r)
        out[(long long)(row0 + mS + r) * ND + n] = acc[t][r];
    }
  }
}

extern "C" void kernel_launch(void* const* d_in, const int* in_sizes, int n_in,
                              void* d_out, int out_size, void* d_ws,
                              size_t ws_size, hipStream_t stream) {
  (void)in_sizes; (void)n_in; (void)out_size; (void)ws_size;

  const int*   a_idx  = (const int*)d_in[0];    // (2, NNZ)
  const float* a_vals = (const float*)d_in[1];  // (NNZ,)
  const float* X      = (const float*)d_in[2];  // (N, B, F)
  const float* W      = (const float*)d_in[3];  // (F, OUT)
  float*       out    = (float*)d_out;          // (N, B, OUT)
  float*       agg    = (float*)d_ws;           // 200000 x 100 f32 = 80 MB

  const int* rows = a_idx;
  const int* cols = a_idx + NNZ_E;

  // 1) zero agg
  const long long n4 = (long long)MROWS * KD / 4;  // 5M float4
  zero_ws_kernel<<<(int)((n4 + 255) / 256), 256, 0, stream>>>((float4*)agg, n4);

  // 2) SpMM scatter with f32 atomics
  const long long total = (long long)NNZ_E * CHUNKS;  // 50M work items
  spmm_scatter_kernel<<<(int)((total + 255) / 256), 256, 0, stream>>>(
      rows, cols, a_vals, (const float4*)X, agg);

  // 3) WMMA GEMM: 12500 M-tiles, 4 waves/block -> 3125 blocks
  gemm_wmma_kernel<<<MROWS / 16 / 4, 128, 0, stream>>>(agg, W, out);
}